// PGFAPoseGuidedMaskBlock_54065048322069
// MI455X (gfx1250) — compile-verified
//
#include <hip/hip_runtime.h>
#include <math.h>

typedef float v2f __attribute__((ext_vector_type(2)));
typedef float v8f __attribute__((ext_vector_type(8)));
typedef unsigned int u32x4 __attribute__((ext_vector_type(4)));
typedef int i32x4 __attribute__((ext_vector_type(4)));
typedef int i32x8 __attribute__((ext_vector_type(8)));

#define BATCH 256
#define CHAN  2048
#define NLM   18
#define HW    192
// Row stride in LDS (floats), padded so WMMA operand reads are bank-conflict-free:
// 196 mod 64 = 4 -> lanes 0-15 hit banks {4L,4L+1}, lanes 16-31 hit {4L+2,4L+3}.
#define LDS_STRIDE 196
#define FEAT_TILE_FLOATS (16 * LDS_STRIDE)  // 12544 B per wave tile
#define MASK_TILE_FLOATS (32 * LDS_STRIDE)  // 25088 B (18 TDM rows + 14 zero rows)
#define WAVES_PER_BLOCK 4
#define CH_PER_BLOCK (WAVES_PER_BLOCK * 16) // 64 channels per block
#define LDS_MASK_BYTE_OFF (WAVES_PER_BLOCK * FEAT_TILE_FLOATS * 4)

// butterfly max via ds_swizzle (group-of-32 encoding: offset = xor<<5 | and_mask).
// IMM is a template param so the builtin sees a constant integer.
template <int IMM>
__device__ __forceinline__ float swz_max(float v) {
    int x = __builtin_amdgcn_ds_swizzle(__float_as_int(v), IMM);
    return fmaxf(v, __int_as_float(x));
}

__global__ void __launch_bounds__(128)
pgfa_pose_guided_mask_kernel(const float* __restrict__ feat,
                             const float* __restrict__ masks,
                             const float* __restrict__ pgf,
                             float* __restrict__ out) {
    extern __shared__ float lds[];
    float* lds_feat = lds;                                          // 4 wave tiles
    float* lds_mask = lds + WAVES_PER_BLOCK * FEAT_TILE_FLOATS;     // shared mask tile

    const int tid     = threadIdx.x;
    const int wave    = tid >> 5;
    const int lane    = tid & 31;
    const int lane_lo = lane & 15;
    const int lane_hi = lane >> 4;

    const int blocks_per_b = CHAN / CH_PER_BLOCK;                   // 32
    const int b          = blockIdx.x / blocks_per_b;
    const int ctile_base = (blockIdx.x % blocks_per_b) * CH_PER_BLOCK;
    const int ctile      = ctile_base + wave * 16;

    // ---- TDM: DMA masks[b] (18 rows x 192 f32) into LDS with 196-float row
    //      stride via iterate mode (1 row per iteration, lds_inc=196, g_inc=192).
    //      Wave-scalar op: only wave 0 issues it. ----
    if (wave == 0) {
        const unsigned long long ga =
            (unsigned long long)(uintptr_t)(masks + (size_t)b * NLM * HW);
        u32x4 g0;
        g0[0] = 1u;                                        // count=1, no gather/restore
        g0[1] = (unsigned)LDS_MASK_BYTE_OFF;               // lds_addr (bytes)
        g0[2] = (unsigned)(ga & 0xFFFFFFFFu);              // global_addr[31:0]
        g0[3] = (unsigned)((ga >> 32) & 0x01FFFFFFu) | (2u << 30);  // addr[56:32], type=2
        i32x8 g1 = {0, 0, 0, 0, 0, 0, 0, 0};
        g1[0] = (2 << 16) | (1 << 19);                     // data_size=4B, iterate_enable=1
        g1[1] = (int)(((unsigned)HW) << 16);               // tensor_dim0[15:0] @ [63:48]
        g1[2] = (int)(((unsigned)NLM) << 16);              // tensor_dim1[15:0] @ [95:80]
        g1[3] = (int)(((unsigned)HW) << 16);               // tile_dim0 @ [127:112]
        g1[4] = 1;                                         // tile_dim1=1 @ [143:128]
        g1[5] = HW;                                        // tensor_dim0_stride[31:0]
        i32x4 g2;
        g2[0] = 0;                                         // tensor_dim2 (unused, 2D)
        g2[1] = LDS_STRIDE;                                // lds_addr_increment (elems)
        g2[2] = HW;                                        // global_addr_increment[31:0]
        g2[3] = (int)(((unsigned)(NLM - 1)) << 16);        // iterate_count=17 @ [127:112]
        i32x4 g3 = {0, 0, 0, 0};
#if __clang_major__ >= 23
        i32x8 g4 = {0, 0, 0, 0, 0, 0, 0, 0};
        __builtin_amdgcn_tensor_load_to_lds(g0, g1, g2, g3, g4, 0);
#else
        __builtin_amdgcn_tensor_load_to_lds(g0, g1, g2, g3, 0);
#endif
        __builtin_amdgcn_s_wait_tensorcnt(0);
    }

    // ---- Zero-fill padded landmark rows 18..31 (read as B-matrix zero columns) ----
    for (int i = tid; i < (32 - NLM) * HW; i += 128) {
        const int n  = NLM + i / HW;
        const int hw = i - (i / HW) * HW;
        lds_mask[n * LDS_STRIDE + hw] = 0.0f;
    }

    // ---- Stage this wave's 16 x 192 feat tile into LDS (float4 coalesced).
    //      Bulk stream (403 MB total) stays on the VMEM path for full HBM rate. ----
    {
        const float* fbase = feat + ((size_t)b * CHAN + ctile) * HW;
        float* ftile = lds_feat + wave * FEAT_TILE_FLOATS;
#pragma unroll
        for (int c = 0; c < 16; ++c) {
            const float4* src = (const float4*)(fbase + c * HW);    // 48 float4 per row
            float4*       dst = (float4*)(ftile + c * LDS_STRIDE);  // 16B aligned rows
            dst[lane] = src[lane];
            if (lane < 16) dst[lane + 32] = src[lane + 32];
        }
    }
    __syncthreads();

    // ---- GEMM: D(16ch x 32lm) += A(16ch x 4hw) * B(4hw x 16lm), 48 k-steps ----
    // A layout: lane = row M=lane_lo, k = kb + 2*lane_hi + {0,1}
    // B layout: lane = col N=lane_lo, k = kb + 2*lane_hi + {0,1}  (same addressing)
    const float* ftile = lds_feat + wave * FEAT_TILE_FLOATS;
    const int a_off  = lane_lo * LDS_STRIDE + 2 * lane_hi;
    const int b0_off = lane_lo * LDS_STRIDE + 2 * lane_hi;
    const int b1_off = (lane_lo + 16) * LDS_STRIDE + 2 * lane_hi;

    v8f acc0 = {};   // landmarks 0..15
    v8f acc1 = {};   // landmarks 16..31 (18..31 are zero columns)

#pragma unroll 4
    for (int kb = 0; kb < HW; kb += 4) {
        v2f a  = *(const v2f*)(ftile    + a_off  + kb);
        v2f b0 = *(const v2f*)(lds_mask + b0_off + kb);
        v2f b1 = *(const v2f*)(lds_mask + b1_off + kb);
        acc0 = __builtin_amdgcn_wmma_f32_16x16x4_f32(
            false, a, false, b0, (short)0, acc0, false, false);
        acc1 = __builtin_amdgcn_wmma_f32_16x16x4_f32(
            false, a, false, b1, (short)0, acc1, false, false);
    }

    // ---- Max over landmark axis, scale by 1/HW, write pg_max half of out ----
    // C/D layout: VGPR v -> M = v + 8*lane_hi, N = lane_lo.
    // XOR-swizzle butterfly (masks 8,4,2,1 stay within each 16-lane half).
    const float inv_hw = 1.0f / (float)HW;
    float* out_b = out + (size_t)b * (2 * CHAN);
#pragma unroll
    for (int v = 0; v < 8; ++v) {
        float val   = acc0[v];                                   // landmarks 0..15
        float extra = (lane_lo < 2) ? acc1[v] : -INFINITY;       // landmarks 16,17 only
        val = fmaxf(val, extra);
        val = swz_max<0x011F>(val);   // xor 8
        val = swz_max<0x009F>(val);   // xor 4
        val = swz_max<0x005F>(val);   // xor 2
        val = swz_max<0x003F>(val);   // xor 1
        if (lane_lo == 0) {
            const int m = v + 8 * lane_hi;
            out_b[CHAN + ctile + m] = val * inv_hw;
        }
    }

    // ---- Copy this block's slice of pool_global_feat into out[:, 0:C] ----
    if (tid < CH_PER_BLOCK) {
        out_b[ctile_base + tid] = pgf[(size_t)b * CHAN + ctile_base + tid];
    }
}

extern "C" void kernel_launch(void* const* d_in, const int* in_sizes, int n_in,
                              void* d_out, int out_size, void* d_ws, size_t ws_size,
                              hipStream_t stream) {
    const float* feat  = (const float*)d_in[0];
    const float* masks = (const float*)d_in[1];
    const float* pgf   = (const float*)d_in[2];
    float* out = (float*)d_out;

    const int grid = BATCH * (CHAN / CH_PER_BLOCK);   // 8192 blocks
    const size_t shmem =
        (WAVES_PER_BLOCK * FEAT_TILE_FLOATS + MASK_TILE_FLOATS) * sizeof(float); // 75264 B

    hipLaunchKernelGGL(pgfa_pose_guided_mask_kernel,
                       dim3(grid), dim3(128), shmem, stream,
                       feat, masks, pgf, out);
}